// PerceptronLayer_54898271977905
// MI455X (gfx1250) — compile-verified
//
#include <hip/hip_runtime.h>

typedef float v2f __attribute__((ext_vector_type(2)));
typedef float v8f __attribute__((ext_vector_type(8)));

#define WMMA_F32(a, b, c) \
  __builtin_amdgcn_wmma_f32_16x16x4_f32(false, (a), false, (b), (short)0, (c), false, false)

constexpr int   DT          = 480;                    // 128 + 192 + 160
constexpr float INV_SQRT128 = 0.08838834764831845f;
constexpr float INV_SQRT64  = 0.125f;
constexpr float INV_SQRT32  = 0.17677669529663687f;
constexpr float CST_SILU    = 1.6765f;                // 1/sqrt(E[silu(z)^2]), z~N(0,1)
constexpr float CST_RELU    = 1.41421356f;            // sqrt(2)

// One block = 256 threads = 8 waves; each wave owns 16 rows.
// All weights staged in LDS once per block (133 KB of the 320 KB WGP budget),
// W0/W1/W2 stored row-PAIR interleaved so a WMMA B-fragment is one ds_load_b64.
__global__ __launch_bounds__(256)
void perceptron_fused(const float* __restrict__ x,
                      const float* __restrict__ W0,
                      const float* __restrict__ b0,
                      const float* __restrict__ W1,
                      const float* __restrict__ W2,
                      float* __restrict__ out)
{
  __shared__ float lds[34016];          // 136,064 bytes
  float* w0p = lds;                     // 28672 floats: W0 (128x224) pair-interleaved
  float* w1p = lds + 28672;             //  4096 floats: W1 (64x64)
  float* w2p = lds + 32768;             //  1024 floats: W2 (32x32)
  float* bb  = lds + 33792;             //   224 floats: b0

  const int tid = threadIdx.x;

  // ---- cooperative weight staging: element (r,c) -> [((r/2)*W + c)*2 + (r&1)] ----
  for (int i = tid; i < 128 * 224; i += 256) {
    int r = i / 224, c = i - r * 224;
    w0p[((((r >> 1) * 224) + c) << 1) | (r & 1)] = W0[i];
  }
  for (int i = tid; i < 64 * 64; i += 256) {
    int r = i >> 6, c = i & 63;
    w1p[((((r >> 1) * 64) + c) << 1) | (r & 1)] = W1[i];
  }
  for (int i = tid; i < 32 * 32; i += 256) {
    int r = i >> 5, c = i & 31;
    w2p[((((r >> 1) * 32) + c) << 1) | (r & 1)] = W2[i];
  }
  if (tid < 224) bb[tid] = b0[tid];

  const int  wave = tid >> 5;
  const int  lane = tid & 31;
  const int  hl   = lane >> 4;   // which K-pair half (WMMA A/B layout)
  const int  lr   = lane & 15;   // A-row / B,C,D-column within tile
  const long row0 = ((long)blockIdx.x * 8 + wave) * 16;
  const float* xrow = x + (row0 + lr) * DT;   // per-lane A-matrix row base

  // warm the cache for this wave's x slab while weights stage (global_prefetch_b8)
  __builtin_prefetch(xrow, 0, 3);
  __builtin_prefetch(xrow + 128, 0, 3);
  __builtin_prefetch(xrow + 256, 0, 3);
  __builtin_prefetch(xrow + 384, 0, 3);

  __syncthreads();

  const v8f vzero = {0.f, 0.f, 0.f, 0.f, 0.f, 0.f, 0.f, 0.f};

  // ================= Phase 1: s = x0 @ W0 (16 x 224, 14 col tiles, K=128) ========
  v8f acc[14];
#pragma unroll
  for (int c = 0; c < 14; ++c) acc[c] = vzero;

  for (int kk = 0; kk < 128; kk += 4) {
    v2f a = *(const v2f*)(xrow + kk + 2 * hl);               // A frag: rows, K pair
    const float* wb = w0p + ((((kk >> 1) + hl) * 224) + lr) * 2;
#pragma unroll
    for (int c = 0; c < 14; ++c) {
      v2f b = *(const v2f*)(wb + 32 * c);                    // B frag (ds_load_b64)
      acc[c] = WMMA_F32(a, b, acc[c]);
    }
  }

  // ---- scalars: silu on col tiles 0..7, streamed straight to out ----
#pragma unroll
  for (int c = 0; c < 8; ++c) {
    float bias = bb[16 * c + lr];
#pragma unroll
    for (int r = 0; r < 8; ++r) {
      float s   = acc[c][r] * INV_SQRT128 + bias;
      float sig = 1.f / (1.f + __expf(-s));
      out[(row0 + r + 8 * hl) * DT + 16 * c + lr] = CST_SILU * s * sig;
    }
  }
  // ---- gates: relu on col tiles 8..13, kept in registers (same layout as v/t C) --
  v8f g[6];
#pragma unroll
  for (int c = 0; c < 6; ++c) {
    float bias = bb[128 + 16 * c + lr];
#pragma unroll
    for (int r = 0; r < 8; ++r) {
      float s = acc[8 + c][r] * INV_SQRT128 + bias;
      g[c][r] = CST_RELU * fmaxf(s, 0.f);
    }
  }

  // ================= Phase 2: v[m] = x1[:,:,m] @ W1 (16 x 64, K=64), gate, store ==
#pragma unroll
  for (int m = 0; m < 3; ++m) {
    v8f va[4];
#pragma unroll
    for (int ct = 0; ct < 4; ++ct) va[ct] = vzero;
    for (int kk = 0; kk < 64; kk += 4) {
      int i0 = kk + 2 * hl;
      v2f a;
      a.x = xrow[128 + 3 * i0 + m];          // x1[n,i,m] = x[n, 128 + 3i + m]
      a.y = xrow[128 + 3 * i0 + 3 + m];
      const float* wb = w1p + ((((kk >> 1) + hl) * 64) + lr) * 2;
#pragma unroll
      for (int ct = 0; ct < 4; ++ct) {
        v2f b = *(const v2f*)(wb + 32 * ct);
        va[ct] = WMMA_F32(a, b, va[ct]);
      }
    }
#pragma unroll
    for (int ct = 0; ct < 4; ++ct)
#pragma unroll
      for (int r = 0; r < 8; ++r)
        out[(row0 + r + 8 * hl) * DT + 128 + 3 * (16 * ct + lr) + m] =
            va[ct][r] * INV_SQRT64 * g[ct][r];   // same C-layout: per-lane gating
  }

  // ================= Phase 3: t[m] = x2[:,:,m] @ W2 (16 x 32, K=32), gate, store ==
#pragma unroll
  for (int m = 0; m < 5; ++m) {
    v8f ta[2];
    ta[0] = vzero; ta[1] = vzero;
    for (int kk = 0; kk < 32; kk += 4) {
      int i0 = kk + 2 * hl;
      v2f a;
      a.x = xrow[320 + 5 * i0 + m];          // x2[n,i,m] = x[n, 320 + 5i + m]
      a.y = xrow[320 + 5 * i0 + 5 + m];
      const float* wb = w2p + ((((kk >> 1) + hl) * 32) + lr) * 2;
#pragma unroll
      for (int ct = 0; ct < 2; ++ct) {
        v2f b = *(const v2f*)(wb + 32 * ct);
        ta[ct] = WMMA_F32(a, b, ta[ct]);
      }
    }
#pragma unroll
    for (int ct = 0; ct < 2; ++ct)
#pragma unroll
      for (int r = 0; r < 8; ++r)
        out[(row0 + r + 8 * hl) * DT + 320 + 5 * (16 * ct + lr) + m] =
            ta[ct][r] * INV_SQRT32 * g[4 + ct][r];
  }
}

extern "C" void kernel_launch(void* const* d_in, const int* in_sizes, int n_in,
                              void* d_out, int out_size, void* d_ws, size_t ws_size,
                              hipStream_t stream) {
  const float* x  = (const float*)d_in[0];
  const float* W0 = (const float*)d_in[1];
  const float* b0 = (const float*)d_in[2];
  const float* W1 = (const float*)d_in[3];
  const float* W2 = (const float*)d_in[4];
  float* out = (float*)d_out;

  const int n      = in_sizes[0] / DT;   // 262144
  const int blocks = n / 128;            // 8 waves x 16 rows per block
  perceptron_fused<<<blocks, 256, 0, stream>>>(x, W0, b0, W1, W2, out);
}